// SignedLayer_13898514170513
// MI455X (gfx1250) — compile-verified
//
#include <hip/hip_runtime.h>
#include <hip/hip_bf16.h>

typedef float v2f __attribute__((ext_vector_type(2)));
typedef float v8f __attribute__((ext_vector_type(8)));

#define NN 100000
#define NE 600000
#define DD 128

// ---------------- degree kernels ----------------

__global__ void deg_init_kernel(float* __restrict__ degp, float* __restrict__ degn) {
    int i = blockIdx.x * blockDim.x + threadIdx.x;
    if (i < NN) { degp[i] = 1.0f; degn[i] = 1.0f; }  // self-loop counts as 1
}

__global__ void deg_count_kernel(const int* __restrict__ dst, float* __restrict__ deg) {
    int i = blockIdx.x * blockDim.x + threadIdx.x;
    if (i < NE) atomicAdd(&deg[dst[i]], 1.0f);
}

__global__ void deg_rsqrt_kernel(float* __restrict__ degp, float* __restrict__ degn) {
    int i = blockIdx.x * blockDim.x + threadIdx.x;
    if (i < NN) { degp[i] = rsqrtf(degp[i]); degn[i] = rsqrtf(degn[i]); }
}

// ---------------- WMMA f32 GEMM: C[n,128] = (A[n,128] (+bin)) @ B[128,128] (+bout) ----------------
// One block = 256 threads = 8 waves; block covers 16 rows, wave w covers cols 16w..16w+15.
// A tile (16x4 f32): lane half h holds K = kb+2h, kb+2h+1 in vgpr pair; row = lane%16.
// B tile (4x16 f32): vgpr i holds row K = kb+2h+i, col = lane%16.
// C tile (16x16 f32): vgpr v holds row v + 8h, col = lane%16.

template <bool ACC, bool HAS_BIN, bool HAS_BOUT>
__global__ __launch_bounds__(256)
void gemm_wmma_kernel(const float* __restrict__ A, const float* __restrict__ B,
                      const float* __restrict__ bin, const float* __restrict__ bout,
                      float* __restrict__ C) {
    const int lane = threadIdx.x & 31;
    const int wave = threadIdx.x >> 5;
    const int m    = lane & 15;   // row-in-tile for A/C, col-in-tile for B
    const int h    = lane >> 4;   // lane half selects K pair
    const int rowbase = blockIdx.x * 16;
    const int colbase = wave * 16;

    const float* __restrict__ Arow = A + (size_t)(rowbase + m) * DD;
    const float* __restrict__ Bcol = B + colbase + m;   // stride DD per K row

    v8f acc = {0.f, 0.f, 0.f, 0.f, 0.f, 0.f, 0.f, 0.f};

#pragma unroll
    for (int kb = 0; kb < DD; kb += 4) {
        const int k0 = kb + 2 * h;
        v2f a = *(const v2f*)(Arow + k0);
        if (HAS_BIN) { a.x += bin[k0]; a.y += bin[k0 + 1]; }
        v2f b;
        b.x = Bcol[(size_t)k0 * DD];
        b.y = Bcol[(size_t)(k0 + 1) * DD];
        acc = __builtin_amdgcn_wmma_f32_16x16x4_f32(
            /*neg_a=*/false, a, /*neg_b=*/false, b,
            /*c_mod=*/(short)0, acc, /*reuse_a=*/false, /*reuse_b=*/false);
    }

#pragma unroll
    for (int v = 0; v < 8; ++v) {
        const int row = rowbase + v + 8 * h;
        const int col = colbase + m;
        float val = acc[v];
        if (HAS_BOUT) val += bout[col];
        float* p = C + (size_t)row * DD + col;
        if (ACC) *p += val; else *p = val;
    }
}

// ---------------- self-loop init: agg = xw * dinv[node]^2 ----------------

__global__ void agg_init_kernel(const float* __restrict__ xw, const float* __restrict__ dinv,
                                float* __restrict__ agg) {
    size_t i = (size_t)blockIdx.x * blockDim.x + threadIdx.x;   // over N*D/4
    size_t idx = i * 4;
    if (idx >= (size_t)NN * DD) return;
    int node = (int)(idx / DD);
    float di = dinv[node];
    float s = di * di;
    const float* src = xw + idx;
    float* dst = agg + idx;
    dst[0] = src[0] * s;
    dst[1] = src[1] * s;
    dst[2] = src[2] * s;
    dst[3] = src[3] * s;
}

// ---------------- edge scatter: one wave per edge, float4 per lane ----------------

__global__ void scatter_kernel(const int* __restrict__ ei, const float* __restrict__ dinv,
                               const float* __restrict__ xw, float* __restrict__ agg) {
    size_t t = (size_t)blockIdx.x * blockDim.x + threadIdx.x;
    int e    = (int)(t >> 5);
    int lane = (int)(t & 31);
    if (e >= NE) return;
    int src = ei[e];
    int dst = ei[NE + e];
    float coef = dinv[src] * dinv[dst];
    const float* sp = xw + (size_t)src * DD + lane * 4;
    float*       dp = agg + (size_t)dst * DD + lane * 4;
    float v0 = sp[0] * coef, v1 = sp[1] * coef, v2 = sp[2] * coef, v3 = sp[3] * coef;
    atomicAdd(dp + 0, v0);
    atomicAdd(dp + 1, v1);
    atomicAdd(dp + 2, v2);
    atomicAdd(dp + 3, v3);
}

// ---------------- launch ----------------

extern "C" void kernel_launch(void* const* d_in, const int* in_sizes, int n_in,
                              void* d_out, int out_size, void* d_ws, size_t ws_size,
                              hipStream_t stream) {
    const float* x      = (const float*)d_in[0];
    const int*   epos   = (const int*)d_in[1];
    const int*   eneg   = (const int*)d_in[2];
    const float* W1     = (const float*)d_in[3];
    const float* b1     = (const float*)d_in[4];
    const float* W2     = (const float*)d_in[5];
    const float* b2     = (const float*)d_in[6];
    const float* weight = (const float*)d_in[7];
    const float* bias   = (const float*)d_in[8];
    float* out = (float*)d_out;

    float* ws    = (float*)d_ws;
    float* xw    = ws;                              // N*D floats
    float* agg   = ws + (size_t)NN * DD;            // N*D floats
    float* dinvp = ws + 2 * (size_t)NN * DD;        // N floats
    float* dinvn = dinvp + NN;                      // N floats

    const int T = 256;
    const int gN  = (NN + T - 1) / T;
    const int gE  = (NE + T - 1) / T;
    const int gND = (int)(((size_t)NN * DD / 4 + T - 1) / T);
    const int gSC = (int)(((size_t)NE * 32 + T - 1) / T);
    const int gGE = NN / 16;   // 6250 row tiles, N divisible by 16

    // degrees (both graphs), deg -> rsqrt(deg) in place
    deg_init_kernel<<<gN, T, 0, stream>>>(dinvp, dinvn);
    deg_count_kernel<<<gE, T, 0, stream>>>(epos + NE, dinvp);
    deg_count_kernel<<<gE, T, 0, stream>>>(eneg + NE, dinvn);
    deg_rsqrt_kernel<<<gN, T, 0, stream>>>(dinvp, dinvn);

    // ---- pass 1: positive graph, writes out = (agg1 + b1) @ weight[:128] ----
    gemm_wmma_kernel<false, false, false><<<gGE, T, 0, stream>>>(x, W1, nullptr, nullptr, xw);
    agg_init_kernel<<<gND, T, 0, stream>>>(xw, dinvp, agg);
    scatter_kernel<<<gSC, T, 0, stream>>>(epos, dinvp, xw, agg);
    gemm_wmma_kernel<false, true, false><<<gGE, T, 0, stream>>>(agg, weight, b1, nullptr, out);

    // ---- pass 2: negative graph, out += (agg2 + b2) @ weight[128:] + bias ----
    gemm_wmma_kernel<false, false, false><<<gGE, T, 0, stream>>>(x, W2, nullptr, nullptr, xw);
    agg_init_kernel<<<gND, T, 0, stream>>>(xw, dinvn, agg);
    scatter_kernel<<<gSC, T, 0, stream>>>(eneg, dinvn, xw, agg);
    gemm_wmma_kernel<true, true, true><<<gGE, T, 0, stream>>>(agg, weight + (size_t)DD * DD, b2, bias, out);
}